// Metasubgraph_semantic_gat_63591285784613
// MI455X (gfx1250) — compile-verified
//
#include <hip/hip_runtime.h>

typedef float v2f __attribute__((ext_vector_type(2)));
typedef float v8f __attribute__((ext_vector_type(8)));

#define NM 50000
#define ND 50000
#define NN (NM + ND)
#define DK 256            // input feature dim (DM == DD)
#define FF 64             // output feature dim
#define SLOPE 0.2f
#define NT_M (NM / 16)    // 3125 row tiles for mirna
#define NT_ALL (NN / 16)  // 6250 row tiles total

// ---------------------------------------------------------------------------
// Kernel 0: init h (=d_out) to 0, emax keys to INT_MIN, denom to 0
// ---------------------------------------------------------------------------
__global__ void k_init(float* __restrict__ h, int* __restrict__ key,
                       float* __restrict__ denom, long nh) {
  long i = (long)blockIdx.x * blockDim.x + threadIdx.x;
  if (i < nh) h[i] = 0.0f;
  if (i < NN) { key[i] = (int)0x80000000; denom[i] = 0.0f; }
}

// ---------------------------------------------------------------------------
// Kernel 1: projection z = [m_sim @ Wm^T ; d_sim @ Wd^T] via f32 WMMA.
// One wave per 16-node tile, computes all 64 output features (4 N-tiles).
// A (16x4 f32): lanes 0-15 = M rows, VGPR v holds K = k + 2*half + v.
// B (4x16 f32): lanes 0-15 = N cols (half selects K pair), same K striping.
// D (16x16 f32): VGPR v -> M = v + 8*half, N = lane&15.
// ---------------------------------------------------------------------------
__global__ __launch_bounds__(256) void k_proj(const float* __restrict__ msim,
                                              const float* __restrict__ dsim,
                                              const float* __restrict__ Wm,
                                              const float* __restrict__ Wd,
                                              float* __restrict__ z) {
  const int wave = blockIdx.x * (blockDim.x >> 5) + (threadIdx.x >> 5);
  if (wave >= NT_ALL) return;               // wave-uniform: EXEC stays all-1s
  const int lane = threadIdx.x & 31;
  const int half = lane >> 4;
  const int l16 = lane & 15;

  const float* X;
  const float* W;
  int rowbase, gbase;
  if (wave < NT_M) {
    X = msim; W = Wm; rowbase = wave * 16; gbase = rowbase;
  } else {
    X = dsim; W = Wd; rowbase = (wave - NT_M) * 16; gbase = NM + rowbase;
  }
  const float* __restrict__ xrow = X + (size_t)(rowbase + l16) * DK;
  const float* __restrict__ w0 = W + (size_t)(0 * 16 + l16) * DK;
  const float* __restrict__ w1 = W + (size_t)(1 * 16 + l16) * DK;
  const float* __restrict__ w2 = W + (size_t)(2 * 16 + l16) * DK;
  const float* __restrict__ w3 = W + (size_t)(3 * 16 + l16) * DK;

  v8f acc0 = {}, acc1 = {}, acc2 = {}, acc3 = {};
#pragma unroll 4
  for (int k = 0; k < DK; k += 4) {
    const int kb = k + 2 * half;
    v2f a  = { xrow[kb], xrow[kb + 1] };
    v2f b0 = { w0[kb],   w0[kb + 1] };
    v2f b1 = { w1[kb],   w1[kb + 1] };
    v2f b2 = { w2[kb],   w2[kb + 1] };
    v2f b3 = { w3[kb],   w3[kb + 1] };
    acc0 = __builtin_amdgcn_wmma_f32_16x16x4_f32(false, a, false, b0, (short)0, acc0, false, false);
    acc1 = __builtin_amdgcn_wmma_f32_16x16x4_f32(false, a, false, b1, (short)0, acc1, false, false);
    acc2 = __builtin_amdgcn_wmma_f32_16x16x4_f32(false, a, false, b2, (short)0, acc2, false, false);
    acc3 = __builtin_amdgcn_wmma_f32_16x16x4_f32(false, a, false, b3, (short)0, acc3, false, false);
  }

#pragma unroll
  for (int v = 0; v < 8; ++v) {
    const int node = gbase + v + 8 * half;
    float* __restrict__ zr = z + (size_t)node * FF + l16;
    zr[0]  = acc0[v];
    zr[16] = acc1[v];
    zr[32] = acc2[v];
    zr[48] = acc3[v];
  }
}

// monotone int encoding of float so signed-int max == float max
__device__ __forceinline__ int f2key(float f) {
  int i = __float_as_int(f);
  return i >= 0 ? i : (i ^ 0x7fffffff);
}
__device__ __forceinline__ float key2f(int k) {
  return __int_as_float(k >= 0 ? k : (k ^ 0x7fffffff));
}

// ---------------------------------------------------------------------------
// Kernel 2: per-edge score e = leaky_relu(dot(z[src], z[dst])); segment max
// via atomicMax on int key. z rows are L2-resident (25.6 MB << 192 MB L2).
// ---------------------------------------------------------------------------
__global__ void k_edge(const int* __restrict__ src, const int* __restrict__ dst,
                       const float* __restrict__ z, float* __restrict__ e,
                       int* __restrict__ key, int E) {
  int k = blockIdx.x * blockDim.x + threadIdx.x;
  if (k >= E) return;
  const int s = src[k], d = dst[k];
  const float4* __restrict__ zs = (const float4*)(z + (size_t)s * FF);
  const float4* __restrict__ zd = (const float4*)(z + (size_t)d * FF);
  float acc = 0.0f;
#pragma unroll
  for (int i = 0; i < FF / 4; ++i) {
    float4 a = zs[i], b = zd[i];
    acc += a.x * b.x + a.y * b.y + a.z * b.z + a.w * b.w;
  }
  const float ev = acc > 0.0f ? acc : SLOPE * acc;
  e[k] = ev;
  atomicMax(&key[d], f2key(ev));
}

// ---------------------------------------------------------------------------
// Kernel 3: ex = exp(e - emax[dst]) (in place over e); denom += ex
// ---------------------------------------------------------------------------
__global__ void k_softexp(const int* __restrict__ dst, float* __restrict__ e,
                          const int* __restrict__ key, float* __restrict__ denom,
                          int E) {
  int k = blockIdx.x * blockDim.x + threadIdx.x;
  if (k >= E) return;
  const int d = dst[k];
  const float mx = key2f(key[d]);            // edge exists -> key was written
  const float v = __expf(e[k] - mx);
  e[k] = v;                                  // same-thread RMW: safe
  unsafeAtomicAdd(&denom[d], v);
}

// ---------------------------------------------------------------------------
// Kernel 4: h[dst] += (ex/denom[dst]) * z[src]. 16 lanes per edge, one float4
// of the 64-float row each; 4 f32 atomics (all L2-resident targets).
// ---------------------------------------------------------------------------
__global__ void k_agg(const int* __restrict__ src, const int* __restrict__ dst,
                      const float* __restrict__ z, const float* __restrict__ ex,
                      const float* __restrict__ denom, float* __restrict__ h,
                      int E) {
  long gid = (long)blockIdx.x * blockDim.x + threadIdx.x;
  int k = (int)(gid >> 4);
  int g = (int)(gid & 15);
  if (k >= E) return;
  const int s = src[k], d = dst[k];
  const float alpha = ex[k] / denom[d];
  const float4 zv = ((const float4*)(z + (size_t)s * FF))[g];
  float* __restrict__ hp = h + (size_t)d * FF + g * 4;
  unsafeAtomicAdd(hp + 0, alpha * zv.x);
  unsafeAtomicAdd(hp + 1, alpha * zv.y);
  unsafeAtomicAdd(hp + 2, alpha * zv.z);
  unsafeAtomicAdd(hp + 3, alpha * zv.w);
}

// ---------------------------------------------------------------------------
// Kernel 5: elu in place on h (=d_out)
// ---------------------------------------------------------------------------
__global__ void k_elu(float* __restrict__ h, long n4) {
  long i = (long)blockIdx.x * blockDim.x + threadIdx.x;
  if (i >= n4) return;
  float4* p = (float4*)h;
  float4 v = p[i];
  v.x = v.x > 0.0f ? v.x : __expf(v.x) - 1.0f;
  v.y = v.y > 0.0f ? v.y : __expf(v.y) - 1.0f;
  v.z = v.z > 0.0f ? v.z : __expf(v.z) - 1.0f;
  v.w = v.w > 0.0f ? v.w : __expf(v.w) - 1.0f;
  p[i] = v;
}

// ---------------------------------------------------------------------------
extern "C" void kernel_launch(void* const* d_in, const int* in_sizes, int n_in,
                              void* d_out, int out_size, void* d_ws, size_t ws_size,
                              hipStream_t stream) {
  const float* msim = (const float*)d_in[0];
  const float* dsim = (const float*)d_in[1];
  const float* Wm   = (const float*)d_in[2];
  const float* Wd   = (const float*)d_in[3];
  const int*   src  = (const int*)d_in[4];
  const int*   dst  = (const int*)d_in[5];
  const int    E    = in_sizes[4];
  float* out = (float*)d_out;

  // workspace layout (floats/ints): z | e | key | denom  (~33 MB)
  float* z     = (float*)d_ws;
  float* e     = z + (size_t)NN * FF;
  int*   key   = (int*)(e + E);
  float* denom = (float*)(key + NN);

  const long nh = (long)NN * FF;  // 6.4M

  k_init<<<(unsigned)((nh + 255) / 256), 256, 0, stream>>>(out, key, denom, nh);
  k_proj<<<(NT_ALL + 7) / 8, 256, 0, stream>>>(msim, dsim, Wm, Wd, z);
  k_edge<<<(E + 255) / 256, 256, 0, stream>>>(src, dst, z, e, key, E);
  k_softexp<<<(E + 255) / 256, 256, 0, stream>>>(dst, e, key, denom, E);
  k_agg<<<(unsigned)(((long)E * 16 + 255) / 256), 256, 0, stream>>>(src, dst, z, e, denom, out, E);
  k_elu<<<(unsigned)((nh / 4 + 255) / 256), 256, 0, stream>>>(out, nh / 4);
}